// HSMax_27393301414036
// MI455X (gfx1250) — compile-verified
//
#include <hip/hip_runtime.h>

#define FANOUT 64
#define BATCHN 65536
#define CWIDTH 4160   // 64 + 64*64

typedef __attribute__((ext_vector_type(2))) float v2f;
typedef __attribute__((ext_vector_type(4))) float v4f;
typedef __attribute__((ext_vector_type(8))) float v8f;

// xor-lane shuffle via ds_swizzle (group-of-32 mode: offset = {xor[14:10], or[9:5], and[4:0]})
template<int MASK>
__device__ __forceinline__ float swz_xor(float x) {
  return __int_as_float(__builtin_amdgcn_ds_swizzle(__float_as_int(x), (MASK << 10) | 0x1F));
}

// D = A(16x4 f32) x ones(4x16) + C : every column of D holds the 4-wide row sums of A
__device__ __forceinline__ v8f wmma_rowsum(v2f a, v8f c) {
  const v2f ones = {1.0f, 1.0f};
  return __builtin_amdgcn_wmma_f32_16x16x4_f32(false, a, false, ones, (short)0, c,
                                               false, false);
}

// One wave handles 16 rows of 64 f32 each; lanes r and r+16 cooperate on row r.
// Returns logZ = max + ln(sum exp(x - max)) for this lane's row.
// red16: per-wave 16-float LDS staging buffer.
__device__ __forceinline__ float softmax_logZ(const float* row64, int half, int r,
                                              volatile float* red16) {
  const float* seg = row64 + half * 32;   // this lane's 32 contiguous columns
  v4f x[8];
#pragma unroll
  for (int i = 0; i < 8; ++i)
    x[i] = *reinterpret_cast<const v4f*>(seg + 4 * i);

  float mx = -3.402823466e38f;
#pragma unroll
  for (int i = 0; i < 8; ++i)
    mx = fmaxf(mx, fmaxf(fmaxf(x[i].x, x[i].y), fmaxf(x[i].z, x[i].w)));
  mx = fmaxf(mx, swz_xor<16>(mx));        // combine the two half-rows -> full row max

  const float L2E = 1.44269504088896340736f;
  v8f a0 = {0.f,0.f,0.f,0.f,0.f,0.f,0.f,0.f};
  v8f a1 = a0, a2 = a0, a3 = a0;          // 4 independent WMMA accumulator chains
#pragma unroll
  for (int i = 0; i < 8; ++i) {
    v2f e0, e1;
    e0.x = __builtin_amdgcn_exp2f((x[i].x - mx) * L2E);
    e0.y = __builtin_amdgcn_exp2f((x[i].y - mx) * L2E);
    e1.x = __builtin_amdgcn_exp2f((x[i].z - mx) * L2E);
    e1.y = __builtin_amdgcn_exp2f((x[i].w - mx) * L2E);
    if ((i & 1) == 0) { a0 = wmma_rowsum(e0, a0); a1 = wmma_rowsum(e1, a1); }
    else              { a2 = wmma_rowsum(e0, a2); a3 = wmma_rowsum(e1, a3); }
  }
  v8f s8 = (a0 + a1) + (a2 + a3);

  // C/D layout: lane<16 holds rows 0..7 in v0..7, lane>=16 holds rows 8..15.
  // Stage through LDS so every lane can fetch its own row's sum.
  if (r == 0) {
    const int base = half * 8;
#pragma unroll
    for (int k = 0; k < 8; ++k) red16[base + k] = s8[k];
  }
  __builtin_amdgcn_wave_barrier();        // DS ops are in-order within a wave
  const float s = red16[r];
  __builtin_amdgcn_wave_barrier();
  return mx + 0.69314718055994530942f * __builtin_amdgcn_logf(s);
}

__global__ __launch_bounds__(256) void hsm_main_kernel(const float* __restrict__ logits,
                                                       const int* __restrict__ nt,
                                                       float* __restrict__ ws) {
  __shared__ float red[8][2][16];
  const int tid  = threadIdx.x;
  const int lane = tid & 31;
  const int w    = tid >> 5;
  const int half = lane >> 4;
  const int r    = lane & 15;
  const int g    = blockIdx.x * 8 + w;    // wave id: 0..4095 (16 rows each)
  const int b    = g * 16 + r;            // this lane's batch row

  const float* row = logits + (size_t)b * CWIDTH;
  const int rt = nt[b];                                   // root target
  const int ct = nt[(size_t)(rt + 1) * BATCHN + b];       // child target (row rt+1)

  // root log-softmax over logits[b, 0:64]
  const float xt      = row[rt];
  const float logZr   = softmax_logZ(row, half, r, &red[w][0][0]);
  const float root_lp = xt - logZr;                       // log-prob at root target

  // child log-softmax over the gathered 64-wide block
  const float* crow  = row + FANOUT + (size_t)rt * FANOUT;
  const float cxt    = crow[ct];
  const float logZc  = softmax_logZ(crow, half, r, &red[w][1][0]);
  const float ce     = logZc - cxt;                       // -log-prob at child target

  // root-loss numerator: butterfly-reduce the 16 row values, one atomic per wave
  float contrib = half ? 0.0f : root_lp;
  contrib += swz_xor<16>(contrib);
  contrib += swz_xor<8>(contrib);
  contrib += swz_xor<4>(contrib);
  contrib += swz_xor<2>(contrib);
  contrib += swz_xor<1>(contrib);
  if (lane == 0) atomicAdd(&ws[128], contrib);

  // per-node CE sums and counts (segment sum by root_t)
  if (half == 0) {
    atomicAdd(&ws[rt], ce);
    atomicAdd(&ws[FANOUT + rt], 1.0f);
  }
}

__global__ void hsm_init_kernel(float* __restrict__ ws) {
  if (threadIdx.x < 160) ws[threadIdx.x] = 0.0f;
}

__global__ void hsm_final_kernel(const float* __restrict__ ws, float* __restrict__ out) {
  const int lane = threadIdx.x;           // one wave of 32
  float nl = 0.0f, pres = 0.0f;
#pragma unroll
  for (int k = 0; k < 2; ++k) {
    const int f = lane + 32 * k;
    const float cnt = ws[FANOUT + f];
    if (cnt > 0.0f) { nl += ws[f] / fmaxf(cnt, 1.0f); pres += 1.0f; }
  }
  nl   += swz_xor<16>(nl);   nl   += swz_xor<8>(nl);   nl   += swz_xor<4>(nl);
  nl   += swz_xor<2>(nl);    nl   += swz_xor<1>(nl);
  pres += swz_xor<16>(pres); pres += swz_xor<8>(pres); pres += swz_xor<4>(pres);
  pres += swz_xor<2>(pres);  pres += swz_xor<1>(pres);
  if (lane == 0) {
    const float loss_root = -ws[128] * (1.0f / (float)BATCHN);
    out[0] = (loss_root + nl) / (1.0f + pres);
  }
}

extern "C" void kernel_launch(void* const* d_in, const int* in_sizes, int n_in,
                              void* d_out, int out_size, void* d_ws, size_t ws_size,
                              hipStream_t stream) {
  (void)in_sizes; (void)n_in; (void)out_size; (void)ws_size;
  const float* logits = (const float*)d_in[0];
  const int*   nt     = (const int*)d_in[1];
  float*       ws     = (float*)d_ws;
  float*       out    = (float*)d_out;

  hsm_init_kernel<<<1, 192, 0, stream>>>(ws);
  hsm_main_kernel<<<BATCHN / 16 / 8, 256, 0, stream>>>(logits, nt, ws);
  hsm_final_kernel<<<1, 32, 0, stream>>>(ws, out);
}